// dtw_loss_17523466568390
// MI455X (gfx1250) — compile-verified
//
#include <hip/hip_runtime.h>

// ---------------------------------------------------------------------------
// Soft-DTW loss (gamma = 1): B=64, N=M=512, D=64, fp32.
// Phase 1: Dist[b,i,j] = |x_i|^2 + |y_j|^2 - 2 x_i.y_j via V_WMMA_F32_16X16X4_F32
// Phase 2: anti-diagonal softmin recurrence, one workgroup per batch, LDS diag
//          buffers, barrier per diagonal.
// Phase 3: mean of R[b][N][M] over batches.
// ---------------------------------------------------------------------------

typedef __attribute__((ext_vector_type(2))) float v2f;
typedef __attribute__((ext_vector_type(8))) float v8f;

#define SDTW_B   64
#define SDTW_T   512
#define SDTW_D   64
#define SDTW_BIG 100000000.0f

// ---------------- Phase 0: row squared norms -------------------------------
__global__ void sdtw_sqnorm_kernel(const float* __restrict__ X,
                                   float* __restrict__ out, int rows) {
    int r = blockIdx.x * blockDim.x + threadIdx.x;
    if (r >= rows) return;
    const float4* p = (const float4*)(X + (size_t)r * SDTW_D);
    float s = 0.0f;
#pragma unroll
    for (int q = 0; q < SDTW_D / 4; ++q) {
        float4 v = p[q];
        s += v.x * v.x + v.y * v.y + v.z * v.z + v.w * v.w;
    }
    out[r] = s;
}

// ---------------- Phase 1: distance matrix via f32 WMMA --------------------
// One wave computes one 16x16 tile of Dist for one batch. 8 waves / block.
// A (16x4 f32) per-lane layout: lane l<16 -> row i0+l, K = k0+{0,1};
//                               lane l>=16 -> row i0+(l-16), K = k0+{2,3}.
// B (4x16) mirrors with lane -> column of Yᵀ, i.e. row of Y: identical fetch.
__global__ void sdtw_dist_wmma_kernel(const float* __restrict__ X,
                                      const float* __restrict__ Y,
                                      const float* __restrict__ x2,
                                      const float* __restrict__ y2,
                                      float* __restrict__ Dist) {
    const int wave = threadIdx.x >> 5;          // 0..7
    const int lane = threadIdx.x & 31;
    const int tile = blockIdx.x * 8 + wave;     // 0..65535
    const int b    = tile >> 10;                // /1024 tiles per batch
    const int t    = tile & 1023;
    const int i0   = (t >> 5) << 4;             // row tile origin
    const int j0   = (t & 31) << 4;             // col tile origin
    const int half = lane >> 4;                 // K sub-pair selector
    const int l    = lane & 15;

    const size_t xrow = ((size_t)b * SDTW_T + i0 + l) * SDTW_D;
    const size_t yrow = ((size_t)b * SDTW_T + j0 + l) * SDTW_D;

    v8f acc = {};
#pragma unroll
    for (int k0 = 0; k0 < SDTW_D; k0 += 4) {
        const int off = k0 + 2 * half;
        const float2 av = *(const float2*)(X + xrow + off);
        const float2 bv = *(const float2*)(Y + yrow + off);
        v2f a; a[0] = av.x; a[1] = av.y;
        v2f bb; bb[0] = bv.x; bb[1] = bv.y;
        acc = __builtin_amdgcn_wmma_f32_16x16x4_f32(
            /*neg_a=*/false, a, /*neg_b=*/false, bb,
            /*c_mod=*/(short)0, acc, /*reuse_a=*/false, /*reuse_b=*/false);
    }

    // Epilogue: C layout -> row = i0 + 8*half + r, col = j0 + l.
    const float yc = y2[(size_t)b * SDTW_T + j0 + l];
#pragma unroll
    for (int r = 0; r < 8; ++r) {
        const int row = i0 + half * 8 + r;
        const float d = x2[(size_t)b * SDTW_T + row] + yc - 2.0f * acc[r];
        Dist[((size_t)b * SDTW_T + row) * SDTW_T + (j0 + l)] = d;
    }
}

// ---------------- Phase 2: serial anti-diagonal recurrence -----------------
// One block (512 threads = 16 wave32) per batch. LDS holds 3 diagonals of
// R indexed by i in [0,512]; rotate each step. gamma = 1.
__global__ void sdtw_recurrence_kernel(const float* __restrict__ Dist,
                                       float* __restrict__ res) {
    __shared__ float rbuf[3][SDTW_T + 1];
    const int b   = blockIdx.x;
    const int tid = threadIdx.x;         // 0..511
    const int i   = tid + 1;             // cell row 1..512

    // diag d=0: only (0,0)=0; diag d=1: all boundary BIG.
    for (int idx = tid; idx <= SDTW_T; idx += SDTW_T) {
        rbuf[0][idx] = (idx == 0) ? 0.0f : SDTW_BIG;
        rbuf[1][idx] = SDTW_BIG;
    }
    __syncthreads();

    float* rA = rbuf[0];   // diag d-2
    float* rB = rbuf[1];   // diag d-1
    float* rC = rbuf[2];   // diag d (being written)

    const float* Drow = Dist + ((size_t)b * SDTW_T + (i - 1)) * SDTW_T;

    for (int d = 2; d <= 2 * SDTW_T; ++d) {
        const int j = d - i;
        float nv = SDTW_BIG;
        if (j >= 1 && j <= SDTW_T) {
            const float cost = Drow[j - 1];
            if (j + 16 <= SDTW_T) __builtin_prefetch(&Drow[j + 15], 0, 1);
            const float a = rB[i - 1];   // R[i-1][j]
            const float c = rB[i];       // R[i][j-1]
            const float e = rA[i - 1];   // R[i-1][j-1]
            const float m = fminf(a, fminf(c, e));
            const float s = __expf(m - a) + __expf(m - c) + __expf(m - e);
            nv = cost + (m - __logf(s));
        }
        rC[i] = nv;
        if (tid == 0) rC[0] = SDTW_BIG;  // row-0 boundary for next steps
        __syncthreads();
        float* tmp = rA; rA = rB; rB = rC; rC = tmp;  // rotate diagonals
    }

    if (tid == 0) res[b] = rB[SDTW_T];   // R[N][M], last diag after rotation
}

// ---------------- Phase 3: mean over batch ---------------------------------
__global__ void sdtw_mean_kernel(const float* __restrict__ res,
                                 float* __restrict__ out) {
    if (threadIdx.x == 0 && blockIdx.x == 0) {
        float s = 0.0f;
        for (int k = 0; k < SDTW_B; ++k) s += res[k];
        out[0] = s / (float)SDTW_B;
    }
}

// ---------------------------------------------------------------------------
extern "C" void kernel_launch(void* const* d_in, const int* in_sizes, int n_in,
                              void* d_out, int out_size, void* d_ws, size_t ws_size,
                              hipStream_t stream) {
    const float* X = (const float*)d_in[0];   // (64,512,64) f32
    const float* Y = (const float*)d_in[1];   // (64,512,64) f32
    float* out = (float*)d_out;               // scalar f32

    // Workspace layout: Dist (64 MB) | x2 | y2 | per-batch results.
    float* Dist = (float*)d_ws;
    float* x2   = Dist + (size_t)SDTW_B * SDTW_T * SDTW_T;
    float* y2   = x2 + (size_t)SDTW_B * SDTW_T;
    float* res  = y2 + (size_t)SDTW_B * SDTW_T;

    const int rows = SDTW_B * SDTW_T;                       // 32768
    sdtw_sqnorm_kernel<<<rows / 256, 256, 0, stream>>>(X, x2, rows);
    sdtw_sqnorm_kernel<<<rows / 256, 256, 0, stream>>>(Y, y2, rows);

    // 65536 16x16 tiles, 8 waves (tiles) per 256-thread block.
    sdtw_dist_wmma_kernel<<<8192, 256, 0, stream>>>(X, Y, x2, y2, Dist);

    sdtw_recurrence_kernel<<<SDTW_B, SDTW_T, 0, stream>>>(Dist, res);

    sdtw_mean_kernel<<<1, 32, 0, stream>>>(res, out);
}